// FComb_79319456023150
// MI455X (gfx1250) — compile-verified
//
#include <hip/hip_runtime.h>

// ---------------------------------------------------------------------------
// Per-voxel MLP (1x1x1 conv chain) on MI455X / gfx1250 using f32 WMMA.
//   h1 = relu(W1x*x + (W1z*z + b1));  h2 = relu(W2*h1 + b2);
//   h3 = relu(W3*h2 + b3);            out = wl.h3 + bl
// 16 spatial points = WMMA M dim; channels = K/N. 48 v_wmma_f32_16x16x4_f32
// per 16-point tile. D->A relayout between layers goes through LDS with an
// interleaved (channel-permuted) column order so both store and load sides
// are 8-byte ops; the K-permutation is absorbed into the B (weight) operands.
// ---------------------------------------------------------------------------

#define C0   32
#define LD   6
#define PDIM 96
#define VOL  (PDIM * PDIM * PDIM)        // 884736 points
#define HSTR 36                           // padded LDS row stride (floats)
#define TILES_PER_WAVE  4
#define WAVES_PER_BLOCK 8
#define POINTS_PER_BLOCK (16 * TILES_PER_WAVE * WAVES_PER_BLOCK)  // 512
#define FLT_BIG 3.402823466e+38f

typedef float v2f __attribute__((ext_vector_type(2)));
typedef float v8f __attribute__((ext_vector_type(8)));

__device__ __forceinline__ v8f wmma_f32(v2f a, v2f b, v8f c) {
  // 8-arg pattern: (neg_a, A, neg_b, B, c_mod, C, reuse_a, reuse_b)
  return __builtin_amdgcn_wmma_f32_16x16x4_f32(false, a, false, b,
                                               (short)0, c, false, false);
}

__device__ __forceinline__ float relu1(float x) {
  // single v_med3_f32 (no NaN-canonicalize double-max)
  return __builtin_amdgcn_fmed3f(x, 0.0f, FLT_BIG);
}

// Physical LDS column s holds logical channel perm(s); bijective on [0,32).
__device__ __forceinline__ int permk(int s) { return (s >> 1) + ((s & 1) << 4); }

__global__ __launch_bounds__(256) void
fcomb_mlp_kernel(const float* __restrict__ fm, const float* __restrict__ z,
                 const float* __restrict__ w1, const float* __restrict__ b1,
                 const float* __restrict__ w2, const float* __restrict__ b2,
                 const float* __restrict__ w3, const float* __restrict__ b3,
                 const float* __restrict__ wl, const float* __restrict__ bl,
                 float* __restrict__ out)
{
  __shared__ float sW1[C0 * (C0 + LD)];
  __shared__ float sW2[C0 * C0];
  __shared__ float sW3[C0 * C0];
  __shared__ float sWl[C0];
  __shared__ float sB1[C0], sB2[C0], sB3[C0];
  __shared__ float sZ[LD];
  __shared__ float sBl[1];
  __shared__ float sHa[WAVES_PER_BLOCK][16 * HSTR];  // per-wave activation stage
  __shared__ float sHb[WAVES_PER_BLOCK][16 * HSTR];
  __shared__ float sO [WAVES_PER_BLOCK][16];         // per-wave output stage

  const int tid = threadIdx.x;
  for (int i = tid; i < C0 * (C0 + LD); i += 256) sW1[i] = w1[i];
  for (int i = tid; i < C0 * C0;        i += 256) sW2[i] = w2[i];
  for (int i = tid; i < C0 * C0;        i += 256) sW3[i] = w3[i];
  if (tid < C0) { sWl[tid] = wl[tid]; sB1[tid] = b1[tid];
                  sB2[tid] = b2[tid]; sB3[tid] = b3[tid]; }
  if (tid < LD) sZ[tid] = z[tid];
  if (tid == 0) sBl[0] = bl[0];
  __syncthreads();

  const int lane = tid & 31;
  const int wave = tid >> 5;
  const int l16  = lane & 15;
  const int hi   = lane >> 4;   // which 16-lane half of the wave

  // Per-lane accumulator-init values (bias / bias+latent), per output half.
  float cinit1[2], cinit2[2], cinit3[2], wlv[2];
#pragma unroll
  for (int h = 0; h < 2; ++h) {
    const int n = l16 + 16 * h;
    float c = sB1[n];
#pragma unroll
    for (int k = 0; k < LD; ++k) c += sW1[n * (C0 + LD) + C0 + k] * sZ[k];
    cinit1[h] = c;
    cinit2[h] = sB2[n];
    cinit3[h] = sB3[n];
    wlv[h]    = sWl[n];
  }
  const float blv = sBl[0];

  // B operands (weights) resident in VGPRs.
  // B layout, 4x16 f32: VGPR r, lane L -> element (K = r + 2*(L>=16), N = L%16).
  // Layer 1 uses logical K order (A comes from global in channel order);
  // layers 2/3 use the interleaved K order matching the LDS column layout.
  v2f Bw1[2][8], Bw2[2][8], Bw3[2][8];
#pragma unroll
  for (int h = 0; h < 2; ++h) {
    const int n = l16 + 16 * h;
#pragma unroll
    for (int j = 0; j < 8; ++j) {
      const int k0 = 4 * j + 2 * hi;
      v2f t;
      t[0] = sW1[n * (C0 + LD) + k0]; t[1] = sW1[n * (C0 + LD) + k0 + 1];
      Bw1[h][j] = t;
      t[0] = sW2[n * C0 + permk(k0)]; t[1] = sW2[n * C0 + permk(k0 + 1)];
      Bw2[h][j] = t;
      t[0] = sW3[n * C0 + permk(k0)]; t[1] = sW3[n * C0 + permk(k0 + 1)];
      Bw3[h][j] = t;
    }
  }

  float* Ha = &sHa[wave][0];
  float* Hb = &sHb[wave][0];

  for (int t = 0; t < TILES_PER_WAVE; ++t) {
    const int pbase = blockIdx.x * POINTS_PER_BLOCK +
                      (wave * TILES_PER_WAVE + t) * 16;

    // ---- Layer 1: A from global feature_map (A: M=point over lanes,
    //      K=channel over VGPRs: reg r, lane L -> (M=L%16, K=r+2*(L>=16))) ----
    v2f a[8];
#pragma unroll
    for (int j = 0; j < 8; ++j) {
      const int k0 = 4 * j + 2 * hi;
      v2f av;
      av[0] = fm[k0 * VOL + pbase + l16];
      av[1] = fm[(k0 + 1) * VOL + pbase + l16];
      a[j] = av;
    }
    v8f acc0, acc1;
#pragma unroll
    for (int r = 0; r < 8; ++r) { acc0[r] = cinit1[0]; acc1[r] = cinit1[1]; }
#pragma unroll
    for (int j = 0; j < 8; ++j) {
      acc0 = wmma_f32(a[j], Bw1[0][j], acc0);
      acc1 = wmma_f32(a[j], Bw1[1][j], acc1);
    }
    // relu + D->A relayout: lane's (chan l16, chan l16+16) pair stored as one
    // b64 at interleaved columns (2*l16, 2*l16+1).
#pragma unroll
    for (int r = 0; r < 8; ++r) {
      v2f p;
      p[0] = relu1(acc0[r]);
      p[1] = relu1(acc1[r]);
      *(v2f*)&Ha[(r + 8 * hi) * HSTR + 2 * l16] = p;
    }

    // ---- Layer 2 (K in interleaved order; matches Bw2) ----
#pragma unroll
    for (int j = 0; j < 8; ++j)
      a[j] = *(const v2f*)&Ha[l16 * HSTR + 4 * j + 2 * hi];
#pragma unroll
    for (int r = 0; r < 8; ++r) { acc0[r] = cinit2[0]; acc1[r] = cinit2[1]; }
#pragma unroll
    for (int j = 0; j < 8; ++j) {
      acc0 = wmma_f32(a[j], Bw2[0][j], acc0);
      acc1 = wmma_f32(a[j], Bw2[1][j], acc1);
    }
#pragma unroll
    for (int r = 0; r < 8; ++r) {
      v2f p;
      p[0] = relu1(acc0[r]);
      p[1] = relu1(acc1[r]);
      *(v2f*)&Hb[(r + 8 * hi) * HSTR + 2 * l16] = p;
    }

    // ---- Layer 3 (K in interleaved order; matches Bw3) ----
#pragma unroll
    for (int j = 0; j < 8; ++j)
      a[j] = *(const v2f*)&Hb[l16 * HSTR + 4 * j + 2 * hi];
#pragma unroll
    for (int r = 0; r < 8; ++r) { acc0[r] = cinit3[0]; acc1[r] = cinit3[1]; }
#pragma unroll
    for (int j = 0; j < 8; ++j) {
      acc0 = wmma_f32(a[j], Bw3[0][j], acc0);
      acc1 = wmma_f32(a[j], Bw3[1][j], acc1);
    }

    // ---- Final linear (NC=1): relu then 32-channel dot, 16-lane tree reduce.
#pragma unroll
    for (int r = 0; r < 8; ++r) {
      const float v0 = relu1(acc0[r]);
      const float v1 = relu1(acc1[r]);
      float s = v0 * wlv[0] + v1 * wlv[1];   // channels (lane) and (lane+16)
      s += __shfl_xor(s, 1);
      s += __shfl_xor(s, 2);
      s += __shfl_xor(s, 4);
      s += __shfl_xor(s, 8);                 // reduce within each 16-lane half
      if (l16 == 0) sO[wave][r + 8 * hi] = s;  // point r (lo half) / 8+r (hi half)
    }
    if (lane < 16) out[pbase + lane] = sO[wave][lane] + blv;
  }
}

extern "C" void kernel_launch(void* const* d_in, const int* in_sizes, int n_in,
                              void* d_out, int out_size, void* d_ws, size_t ws_size,
                              hipStream_t stream) {
  (void)in_sizes; (void)n_in; (void)out_size; (void)d_ws; (void)ws_size;
  const float* fm = (const float*)d_in[0];
  const float* z  = (const float*)d_in[1];
  const float* w1 = (const float*)d_in[2];
  const float* b1 = (const float*)d_in[3];
  const float* w2 = (const float*)d_in[4];
  const float* b2 = (const float*)d_in[5];
  const float* w3 = (const float*)d_in[6];
  const float* b3 = (const float*)d_in[7];
  const float* wl = (const float*)d_in[8];
  const float* bl = (const float*)d_in[9];
  float* out = (float*)d_out;

  dim3 grid(VOL / POINTS_PER_BLOCK);   // 1728 blocks
  dim3 block(256);                     // 8 wave32 waves
  fcomb_mlp_kernel<<<grid, block, 0, stream>>>(fm, z, w1, b1, w2, b2,
                                               w3, b3, wl, bl, out);
}